// ExpHydro_29317446762563
// MI455X (gfx1250) — compile-verified
//
#include <hip/hip_runtime.h>
#include <hip/hip_bf16.h>
#include <stdint.h>

#define B_DIM 8192
#define T_DIM 3650
#define BLK 32                        // one wave32 per block: no barriers needed
#define NBLK (B_DIM / BLK)            // 256 blocks -> 256 waves spread over the chip
#define CHUNK 16                      // timesteps per LDS tile
#define NCHUNK (T_DIM / CHUNK)        // 228 full chunks
#define TAIL_START (NCHUNK * CHUNK)   // 3648
#define ROW_DATA_F 48                 // 16 steps * 3 floats = 192B per row
#define ROW_F 51                      // TDM pad: +1 dword per 16 dwords -> stride 51, gcd(51,64)=1
#define BUF_F (BLK * ROW_F)           // 1632 floats per buffer
#define BUF_B (BUF_F * 4)             // 6528 bytes per buffer
#define NBUF 3                        // triple buffer, prefetch 2 chunks ahead
#define X_ROW_F (T_DIM * 3)           // 10950 floats between batch rows
#define X_ROW_BYTES (X_ROW_F * 4)     // 43800 B

typedef uint32_t u32x4 __attribute__((ext_vector_type(4)));
typedef uint32_t u32x8 __attribute__((ext_vector_type(8)));

__device__ __forceinline__ float sigmoidf_fast(float x) {
  // 1/(1+e^-x) = rcp(1 + 2^(-x*log2e))
  return __builtin_amdgcn_rcpf(1.0f + __builtin_amdgcn_exp2f(-1.4426950408889634f * x));
}

// step(v) = (tanh(5v)+1)/2 == sigmoid(10v)
__device__ __forceinline__ float stepf_(float v) {
#if __has_builtin(__builtin_amdgcn_tanhf)
  return __builtin_amdgcn_tanhf(5.0f * v) * 0.5f + 0.5f;   // v_tanh_f32
#else
  return __builtin_amdgcn_rcpf(1.0f + __builtin_amdgcn_exp2f(-14.426950408889634f * v));
#endif
}

// One TDM descriptor issue: DMA a (32 rows x 192B, row-stride 43800B) tile into LDS
// with hardware padding (1 dword after every 16) -> LDS row stride 51 dwords.
__device__ __forceinline__ void tdm_issue(uint64_t gaddr, uint32_t lds_byte,
                                          const u32x8& g1) {
  u32x4 g0;
  g0.x = 1u;                                                   // count=1, user D#
  g0.y = lds_byte;                                             // lds_addr
  g0.z = (uint32_t)gaddr;                                      // global_addr[31:0]
  g0.w = ((uint32_t)(gaddr >> 32) & 0x01FFFFFFu) | 0x80000000u; // addr[56:32] | type=2
  asm volatile("tensor_load_to_lds %0, %1" :: "s"(g0), "s"(g1) : "memory");
}

__device__ __forceinline__ int padpos(int idx) { return idx + (idx >> 4); }

__device__ __forceinline__ float hydro_step(float P, float Tt, float Pe,
    float& S1, float& S2,
    float Smax_, float Qmax_, float Df_, float Tmax_, float Tmin_,
    float invSmax, float k2) {
  float s2a   = stepf_(S2);
  float diff  = S2 - Smax_;
  float above = stepf_(diff);
  float below = 1.0f - above;                       // step(Smax-S2) == 1-step(S2-Smax)
  float e     = __builtin_amdgcn_exp2f(k2 * diff);  // exp(-f*(Smax-S2))
  float Qb    = s2a * Qmax_ * fmaf(below, e, above);
  float Qs    = s2a * above * diff;
  float Q     = Qb + Qs;
  float mTT   = Tt - Tmax_;
  float M     = stepf_(mTT) * stepf_(S1) * fminf(S1, Df_ * mTT);
  float snow  = stepf_(Tmin_ - Tt);
  float Ps    = snow * P;
  float Pr    = P - Ps;                             // step(Tt-Tmin)*P == (1-snow)*P
  float ET    = s2a * Pe * fmaf(below * S2, invSmax, above);
  S1 = S1 + (Ps - M);
  S2 = S2 + (Pr + M - ET - Q);
  return Q;
}

__global__ __launch_bounds__(BLK) void exphydro_scan_kernel(
    const float* __restrict__ x,
    const float* __restrict__ fp,   const float* __restrict__ Smaxp,
    const float* __restrict__ Qmaxp,const float* __restrict__ Dfp,
    const float* __restrict__ Tmaxp,const float* __restrict__ Tminp,
    float* __restrict__ out) {
  __shared__ float sh[NBUF * BUF_F];   // 19584 bytes, triple-buffered TDM tile

  const int tid = threadIdx.x;
  const int b   = blockIdx.x * BLK + tid;

  // parameter transforms (uniform; computed once per thread)
  const float f_    = sigmoidf_fast(fp[0])    * 0.1f;
  const float Smax_ = sigmoidf_fast(Smaxp[0]) * 1400.0f + 100.0f;
  const float Qmax_ = sigmoidf_fast(Qmaxp[0]) * 50.0f   + 10.0f;
  const float Df_   = sigmoidf_fast(Dfp[0])   * 5.0f    + 0.01f;
  const float Tmax_ = sigmoidf_fast(Tmaxp[0]) * 3.0f;
  const float Tmin_ = sigmoidf_fast(Tminp[0]) * -3.0f;
  const float invSmax = __builtin_amdgcn_rcpf(Smax_);
  const float k2 = f_ * 1.4426950408889634f;  // exp(f*(S2-Smax)) = exp2(k2*(S2-Smax))

  // D# group1: constant tile/tensor geometry (all wave-uniform -> SGPRs)
  u32x8 g1;
  g1.s0 = (2u << 16)        // data_size = 4 bytes
        | (1u << 20)        // pad_enable
        | (3u << 22)        // pad_interval: after every 16 dwords
        | (0u << 25);       // pad_amount: 1 dword         (workgroup_mask[15:0] = 0)
  g1.s1 = (uint32_t)ROW_DATA_F << 16;   // tensor_dim0[15:0] in [31:16]  (=48)
  g1.s2 = (uint32_t)BLK << 16;          // tensor_dim1[15:0] in [31:16]  (=32)
  g1.s3 = (uint32_t)ROW_DATA_F << 16;   // tile_dim0 in [31:16]          (=48)
  g1.s4 = (uint32_t)BLK;                // tile_dim1 in [15:0]           (=32)
  g1.s5 = (uint32_t)X_ROW_F;            // tensor_dim0_stride low 32     (=10950)
  g1.s6 = 0u;                           // stride high 16 | tensor_dim1_stride low
  g1.s7 = 0u;

  const uint64_t xblk = (uint64_t)(uintptr_t)x + (uint64_t)blockIdx.x * BLK * X_ROW_BYTES;
  const uint32_t shbase = (uint32_t)(uintptr_t)&sh[0];  // LDS offset = low 32 bits

  float S1 = 0.0f, S2 = 0.0f;

  // prime the pipeline two chunks deep (one TDM op per chunk)
  tdm_issue(xblk + 0ull * (CHUNK * 12), shbase + 0u * BUF_B, g1);
  tdm_issue(xblk + 1ull * (CHUNK * 12), shbase + 1u * BUF_B, g1);

  int buf = 0;            // buffer holding chunk c
  for (int c = 0; c < NCHUNK; ++c) {
    if (c + 2 < NCHUNK) {
      int nbuf = buf + 2; if (nbuf >= NBUF) nbuf -= NBUF;   // == (c+2) % NBUF
      tdm_issue(xblk + (uint64_t)(c + 2) * (CHUNK * 12),
                shbase + (uint32_t)nbuf * BUF_B, g1);
      // <=2 outstanding: chunks c+1,c+2 in flight, chunk c landed (in-order per wave)
      asm volatile("s_wait_tensorcnt 2" ::: "memory");
    } else if (c + 1 < NCHUNK) {
      asm volatile("s_wait_tensorcnt 1" ::: "memory");      // chunk c landed
    } else {
      asm volatile("s_wait_tensorcnt 0" ::: "memory");      // last chunk landed
    }
    // single wave per workgroup: producer == consumer, no barrier needed

    const float* rowp = &sh[buf * BUF_F + tid * ROW_F];
    float2 qv[CHUNK / 2];
    float* q = (float*)qv;
#pragma unroll
    for (int k = 0; k < CHUNK; ++k) {
      float P  = rowp[padpos(k * 3 + 0)];
      float Tt = rowp[padpos(k * 3 + 1)];
      float Pe = rowp[padpos(k * 3 + 2)];
      q[k] = hydro_step(P, Tt, Pe, S1, S2, Smax_, Qmax_, Df_, Tmax_, Tmin_, invSmax, k2);
    }

    // out[b*T + t]: base 8B-aligned (b*14600, 64*c) -> float2 stores
    float2* o = (float2*)(out + (size_t)b * T_DIM + (size_t)(c * CHUNK));
#pragma unroll
    for (int k = 0; k < CHUNK / 2; ++k) o[k] = qv[k];

    if (++buf >= NBUF) buf = 0;
  }

  // tail: 3650 - 3648 = 2 steps, plain loads/stores
  const float* xrow = (const float*)((const char*)x + (size_t)b * X_ROW_BYTES);
  for (int t = TAIL_START; t < T_DIM; ++t) {
    float P  = xrow[t * 3 + 0];
    float Tt = xrow[t * 3 + 1];
    float Pe = xrow[t * 3 + 2];
    out[(size_t)b * T_DIM + t] =
        hydro_step(P, Tt, Pe, S1, S2, Smax_, Qmax_, Df_, Tmax_, Tmin_, invSmax, k2);
  }
}

extern "C" void kernel_launch(void* const* d_in, const int* in_sizes, int n_in,
                              void* d_out, int out_size, void* d_ws, size_t ws_size,
                              hipStream_t stream) {
  const float* x     = (const float*)d_in[0];
  const float* f     = (const float*)d_in[1];
  const float* Smax  = (const float*)d_in[2];
  const float* Qmax  = (const float*)d_in[3];
  const float* Df    = (const float*)d_in[4];
  const float* Tmax  = (const float*)d_in[5];
  const float* Tmin  = (const float*)d_in[6];
  float* out         = (float*)d_out;

  exphydro_scan_kernel<<<dim3(NBLK), dim3(BLK), 0, stream>>>(
      x, f, Smax, Qmax, Df, Tmax, Tmin, out);
}